// TgcnCell_3547642987461
// MI455X (gfx1250) — compile-verified
//
#include <hip/hip_runtime.h>
#include <hip/hip_bf16.h>

// ---------------- problem constants ----------------
#define NN      10000     // nodes
#define BB      32        // batch
#define IN_SZ   2
#define UNITS   64
#define F_RAW   66        // IN_SZ + UNITS
#define FPAD    68        // pad K to multiple of 4 for 16x16x4 f32 WMMA
#define ROWVEC  (BB * FPAD)          // 2176 floats per node row in x0/x1
#define CHUNKS  (ROWVEC / 4)         // 544 float4 chunks per row (divisible by 32)
#define NROWS   (BB * NN)            // 320000 GEMM rows
#define VAL_PER_B (NN * 2 * UNITS)   // 1,280,000 (flat per-batch length of `value`)
#define HALF_PER_B (NN * UNITS)      // 640,000
#define SCAN_T  1024
#define SCAN_PER ((NN + SCAN_T - 1) / SCAN_T)   // 10

typedef float v2f __attribute__((ext_vector_type(2)));
typedef float v8f __attribute__((ext_vector_type(8)));

// ---------------- feature build: x0[n][b][f] (FPAD padded, zero tail) ----------------
// use_r: multiply state features by r = value[:, :HALF_PER_B] (flat split, faithful to ref)
__global__ void build_x0_kernel(const float* __restrict__ inputs,
                                const float* __restrict__ state,
                                const float* __restrict__ value,
                                int use_r,
                                float* __restrict__ x0)
{
    size_t idx = (size_t)blockIdx.x * blockDim.x + threadIdx.x;
    const size_t total = (size_t)NN * BB * FPAD;
    if (idx >= total) return;
    int f = (int)(idx % FPAD);
    size_t nb = idx / FPAD;
    int b = (int)(nb % BB);
    int n = (int)(nb / BB);
    float v;
    if (f < IN_SZ) {
        v = inputs[(size_t)b * (NN * IN_SZ) + (size_t)n * IN_SZ + f];
    } else if (f < F_RAW) {
        int o = f - IN_SZ;
        float s = state[(size_t)b * HALF_PER_B + (size_t)n * UNITS + o];
        if (use_r) s *= value[(size_t)b * VAL_PER_B + (size_t)n * UNITS + o];
        v = s;
    } else {
        v = 0.0f;
    }
    x0[idx] = v;
}

// pad + K-pair interleave: wp[(k>>1)*(2*out) + o*2 + (k&1)] = w[k][o]  (zeros for k>=66)
// -> a WMMA B-fragment (w[kk][col], w[kk+1][col]) is one contiguous 8-byte word.
__global__ void pad_w_kernel(const float* __restrict__ w, float* __restrict__ wp, int out)
{
    int idx = blockIdx.x * blockDim.x + threadIdx.x;
    if (idx >= FPAD * out) return;
    int k = idx / out;
    int o = idx % out;
    float v = (k < F_RAW) ? w[k * out + o] : 0.0f;
    wp[(k >> 1) * (2 * out) + o * 2 + (k & 1)] = v;
}

// ---------------- CSR build (once per launch; replaces 1.4G float atomics) ----------------
__global__ void zero_int_kernel(int* __restrict__ p, int n)
{
    int i = blockIdx.x * blockDim.x + threadIdx.x;
    if (i < n) p[i] = 0;
}

__global__ void count_kernel(const int* __restrict__ m_rows, int* __restrict__ counts, int nnz)
{
    int i = blockIdx.x * blockDim.x + threadIdx.x;
    if (i < nnz) atomicAdd(&counts[m_rows[i]], 1);
}

// single-block exclusive scan over NN counts -> row_start[NN+1], cursor copy
__global__ void scan_kernel(const int* __restrict__ counts,
                            int* __restrict__ row_start,
                            int* __restrict__ cursor)
{
    __shared__ int part[SCAN_T];
    int tid = threadIdx.x;
    int base = tid * SCAN_PER;
    int local[SCAN_PER];
    int sum = 0;
    #pragma unroll
    for (int i = 0; i < SCAN_PER; ++i) {
        int idx = base + i;
        int v = (idx < NN) ? counts[idx] : 0;
        local[i] = sum;          // exclusive within this thread's span
        sum += v;
    }
    part[tid] = sum;
    __syncthreads();
    // Hillis-Steele inclusive scan of the 1024 partials (LDS)
    for (int off = 1; off < SCAN_T; off <<= 1) {
        int v = part[tid];
        int add = (tid >= off) ? part[tid - off] : 0;
        __syncthreads();
        part[tid] = v + add;
        __syncthreads();
    }
    int excl = (tid == 0) ? 0 : part[tid - 1];
    #pragma unroll
    for (int i = 0; i < SCAN_PER; ++i) {
        int idx = base + i;
        if (idx < NN) {
            int v = excl + local[i];
            row_start[idx] = v;
            cursor[idx]    = v;
        }
    }
    if (tid == SCAN_T - 1) row_start[NN] = part[SCAN_T - 1];
}

__global__ void scatter_kernel(const int*   __restrict__ m_rows,
                               const int*   __restrict__ m_cols,
                               const float* __restrict__ m_vals,
                               int*   __restrict__ cursor,
                               int*   __restrict__ ecol,
                               float* __restrict__ eval,
                               int nnz)
{
    int i = blockIdx.x * blockDim.x + threadIdx.x;
    if (i >= nnz) return;
    int pos = atomicAdd(&cursor[m_rows[i]], 1);
    ecol[pos] = m_cols[i];
    eval[pos] = m_vals[i];
}

// ---------------- atomic-free CSR SpMM: x1[r][j] = sum_e v_e * x0[c_e][j] ----------------
// thread = one float4 chunk of one row; CHUNKS=544 is a multiple of 32, so every
// wave is row-uniform (uniform loop bounds, broadcastable edge-list loads).
// x0 (87 MB) is L2-resident on MI455X (192 MB L2): the random gather stays in L2.
__global__ void spmm_csr_kernel(const int*   __restrict__ row_start,
                                const int*   __restrict__ ecol,
                                const float* __restrict__ eval,
                                const float* __restrict__ x0,
                                float*       __restrict__ x1)
{
    int idx = blockIdx.x * blockDim.x + threadIdx.x;   // NN*CHUNKS total, exact grid
    int row = idx / CHUNKS;
    int j4  = idx % CHUNKS;
    int beg = row_start[row];
    int end = row_start[row + 1];
    float4 acc = {0.f, 0.f, 0.f, 0.f};
    for (int k = beg; k < end; ++k) {
        int   c = ecol[k];
        float v = eval[k];
        const float4 s = ((const float4*)(x0 + (size_t)c * ROWVEC))[j4];
        acc.x += v * s.x;
        acc.y += v * s.y;
        acc.z += v * s.z;
        acc.w += v * s.w;
    }
    ((float4*)(x1 + (size_t)row * ROWVEC))[j4] = acc;
}

// ---------------- WMMA f32 16x16x4 GEMMs, interleaved weights staged in LDS ----------------
// A: x1 rows (row = b*NN + n  ->  x1 + (n*BB + b)*FPAD), K = FPAD (zero padded)
// Each wave computes a 16(M) x 64(N) macro-tile with 4 accumulators.
// LDS weight layout: sw[(kk>>1)*(2*OUT) + col*2 + {0,1}] -> B fragment = 1x ds_load_b64.

// GEMM1: OUT=128, epilogue = bias + sigmoid -> value[b][n][128] (flat b-major)
__global__ void gemm1_kernel(const float* __restrict__ x1,
                             const float* __restrict__ w,     // [FPAD/2][128][2] interleaved
                             const float* __restrict__ bias,  // [128]
                             float*       __restrict__ value) // [BB][NN*128]
{
    __shared__ float sw[FPAD * 128];                 // 34,816 B of 320 KB/WGP
    for (int i = threadIdx.x; i < FPAD * 128; i += blockDim.x) sw[i] = w[i];
    __syncthreads();

    int wave = (int)((blockIdx.x * blockDim.x + threadIdx.x) >> 5);
    int lane = threadIdx.x & 31;
    int rowTile = wave >> 1;            // 20000 row tiles
    int colHalf = wave & 1;             // 0: cols 0..63, 1: cols 64..127
    int row0 = rowTile * 16;
    int m    = lane & 15;
    int kh   = (lane >> 4) * 2;         // K sub-select per half-wave
    int row  = row0 + m;
    int b    = row / NN;
    int n    = row % NN;
    const float* arow = x1 + ((size_t)n * BB + b) * FPAD;
    int ncol = colHalf * 64 + (lane & 15);

    v8f acc0 = {}, acc1 = {}, acc2 = {}, acc3 = {};
    for (int k = 0; k < FPAD; k += 4) {
        v2f a = *(const v2f*)(arow + k + kh);                    // contiguous b64
        const float* wk = sw + ((k + kh) >> 1) * 256;            // interleaved K-pair row
        v2f b0 = *(const v2f*)(wk + (ncol +  0) * 2);
        v2f b1 = *(const v2f*)(wk + (ncol + 16) * 2);
        v2f b2 = *(const v2f*)(wk + (ncol + 32) * 2);
        v2f b3 = *(const v2f*)(wk + (ncol + 48) * 2);
        acc0 = __builtin_amdgcn_wmma_f32_16x16x4_f32(false, a, false, b0, (short)0, acc0, false, false);
        acc1 = __builtin_amdgcn_wmma_f32_16x16x4_f32(false, a, false, b1, (short)0, acc1, false, false);
        acc2 = __builtin_amdgcn_wmma_f32_16x16x4_f32(false, a, false, b2, (short)0, acc2, false, false);
        acc3 = __builtin_amdgcn_wmma_f32_16x16x4_f32(false, a, false, b3, (short)0, acc3, false, false);
    }

    int mbase = (lane >> 4) * 8;
    #pragma unroll
    for (int t = 0; t < 4; ++t) {
        v8f acc = (t == 0) ? acc0 : (t == 1) ? acc1 : (t == 2) ? acc2 : acc3;
        int col = colHalf * 64 + t * 16 + (lane & 15);
        float bv = bias[col];
        #pragma unroll
        for (int rI = 0; rI < 8; ++rI) {
            int grow = row0 + mbase + rI;
            int gb = grow / NN, gn = grow % NN;
            float x = acc[rI] + bv;
            x = 1.0f / (1.0f + __expf(-x));                       // sigmoid
            value[(size_t)gb * VAL_PER_B + (size_t)gn * 128 + col] = x;
        }
    }
}

// GEMM2: OUT=64, epilogue = bias + relu, fused gate: out = u*state + (1-u)*c
__global__ void gemm2_kernel(const float* __restrict__ x1,
                             const float* __restrict__ w,     // [FPAD/2][64][2] interleaved
                             const float* __restrict__ bias,  // [64]
                             const float* __restrict__ value, // [BB][NN*128] (for u)
                             const float* __restrict__ state, // [BB][NN*64]
                             float*       __restrict__ out)   // [BB][NN*64]
{
    __shared__ float sw[FPAD * 64];                  // 17,408 B
    for (int i = threadIdx.x; i < FPAD * 64; i += blockDim.x) sw[i] = w[i];
    __syncthreads();

    int wave = (int)((blockIdx.x * blockDim.x + threadIdx.x) >> 5);
    int lane = threadIdx.x & 31;
    int row0 = wave * 16;                // one wave covers all 64 output cols
    int m    = lane & 15;
    int kh   = (lane >> 4) * 2;
    int row  = row0 + m;
    int b    = row / NN;
    int n    = row % NN;
    const float* arow = x1 + ((size_t)n * BB + b) * FPAD;
    int ncol = lane & 15;

    v8f acc0 = {}, acc1 = {}, acc2 = {}, acc3 = {};
    for (int k = 0; k < FPAD; k += 4) {
        v2f a = *(const v2f*)(arow + k + kh);
        const float* wk = sw + ((k + kh) >> 1) * 128;
        v2f b0 = *(const v2f*)(wk + (ncol +  0) * 2);
        v2f b1 = *(const v2f*)(wk + (ncol + 16) * 2);
        v2f b2 = *(const v2f*)(wk + (ncol + 32) * 2);
        v2f b3 = *(const v2f*)(wk + (ncol + 48) * 2);
        acc0 = __builtin_amdgcn_wmma_f32_16x16x4_f32(false, a, false, b0, (short)0, acc0, false, false);
        acc1 = __builtin_amdgcn_wmma_f32_16x16x4_f32(false, a, false, b1, (short)0, acc1, false, false);
        acc2 = __builtin_amdgcn_wmma_f32_16x16x4_f32(false, a, false, b2, (short)0, acc2, false, false);
        acc3 = __builtin_amdgcn_wmma_f32_16x16x4_f32(false, a, false, b3, (short)0, acc3, false, false);
    }

    int mbase = (lane >> 4) * 8;
    #pragma unroll
    for (int t = 0; t < 4; ++t) {
        v8f acc = (t == 0) ? acc0 : (t == 1) ? acc1 : (t == 2) ? acc2 : acc3;
        int col = t * 16 + (lane & 15);
        float bv = bias[col];
        #pragma unroll
        for (int rI = 0; rI < 8; ++rI) {
            int grow = row0 + mbase + rI;
            int gb = grow / NN, gn = grow % NN;
            float c = fmaxf(acc[rI] + bv, 0.0f);                  // relu
            size_t j = (size_t)gn * UNITS + col;                  // flat per-batch [N*64]
            float u = value[(size_t)gb * VAL_PER_B + HALF_PER_B + j];
            float s = state[(size_t)gb * HALF_PER_B + j];
            out[(size_t)gb * HALF_PER_B + j] = u * s + (1.0f - u) * c;
        }
    }
}

// ---------------- launcher ----------------
extern "C" void kernel_launch(void* const* d_in, const int* in_sizes, int n_in,
                              void* d_out, int out_size, void* d_ws, size_t ws_size,
                              hipStream_t stream) {
    const float* inputs = (const float*)d_in[0];   // [B, N*2]
    const float* state  = (const float*)d_in[1];   // [B, N*64]
    const int*   m_rows = (const int*)d_in[2];     // [nnz]
    const int*   m_cols = (const int*)d_in[3];     // [nnz]
    const float* m_vals = (const float*)d_in[4];   // [nnz]
    const float* w1     = (const float*)d_in[5];   // [66,128]
    const float* b1     = (const float*)d_in[6];   // [128]
    const float* w2     = (const float*)d_in[7];   // [66,64]
    const float* b2     = (const float*)d_in[8];   // [64]
    float* out = (float*)d_out;
    const int nnz = in_sizes[2];

    // workspace carve-up
    float* ws    = (float*)d_ws;
    const size_t X0_SZ  = (size_t)NN * BB * FPAD;     // 21,760,000 floats
    float* x0    = ws;
    float* x1    = x0 + X0_SZ;
    float* value = x1 + X0_SZ;                        // BB*NN*128 = 40,960,000
    float* w1p   = value + (size_t)BB * VAL_PER_B;
    float* w2p   = w1p + (size_t)FPAD * 128;
    float* evalp = w2p + (size_t)FPAD * 64;           // [nnz] sorted edge values
    int*   ip    = (int*)(evalp + (size_t)nnz);
    int*   row_start = ip;                            // [NN+1]
    int*   cursor    = row_start + (NN + 1);          // [NN]  (counts, then cursors)
    int*   ecol      = cursor + NN;                   // [nnz] sorted edge cols

    const int TPB = 256;
    const int gBuild = (int)((X0_SZ + TPB - 1) / TPB);        // 85000
    const int gNnz   = (nnz + TPB - 1) / TPB;
    const int gSpmm  = (NN * CHUNKS) / TPB;                   // 21250, exact
    const int gGemm1 = (NROWS / 16) * 2 * 32 / TPB;           // 5000, exact
    const int gGemm2 = (NROWS / 16) * 32 / TPB;               // 2500, exact

    // pad + interleave weights
    pad_w_kernel<<<(FPAD * 128 + TPB - 1) / TPB, TPB, 0, stream>>>(w1, w1p, 128);
    pad_w_kernel<<<(FPAD * 64  + TPB - 1) / TPB, TPB, 0, stream>>>(w2, w2p, 64);

    // ---- CSR build (counts in cursor, scan -> row_start + fresh cursor) ----
    zero_int_kernel<<<(NN + TPB - 1) / TPB, TPB, 0, stream>>>(cursor, NN);
    count_kernel   <<<gNnz, TPB, 0, stream>>>(m_rows, cursor, nnz);
    scan_kernel    <<<1, SCAN_T, 0, stream>>>(cursor, row_start, cursor);
    scatter_kernel <<<gNnz, TPB, 0, stream>>>(m_rows, m_cols, m_vals, cursor, ecol, evalp, nnz);

    // ---- pass 1: gc(inputs, state) -> sigmoid -> value ----
    build_x0_kernel<<<gBuild, TPB, 0, stream>>>(inputs, state, value, 0, x0);
    spmm_csr_kernel<<<gSpmm,  TPB, 0, stream>>>(row_start, ecol, evalp, x0, x1);
    gemm1_kernel   <<<gGemm1, TPB, 0, stream>>>(x1, w1p, b1, value);

    // ---- pass 2: gc(inputs, r*state) -> relu -> fused gate -> out ----
    build_x0_kernel<<<gBuild, TPB, 0, stream>>>(inputs, state, value, 1, x0);
    spmm_csr_kernel<<<gSpmm,  TPB, 0, stream>>>(row_start, ecol, evalp, x0, x1);
    gemm2_kernel   <<<gGemm2, TPB, 0, stream>>>(x1, w2p, b2, value, state, out);
}